// MultiHeadAttention_7584912245188
// MI455X (gfx1250) — compile-verified
//
#include <hip/hip_runtime.h>
#include <hip/hip_bf16.h>

// MultiHeadAttention (no softmax) — algebraically fused:
//   out = ( q @ (scale * k^T v) ) @ Wff + bff,  qkv = x @ Wqkv + bqkv
// All matrix math on V_WMMA_F32_16X16X4_F32 (fp32, matches reference numerics).

typedef float v2f __attribute__((ext_vector_type(2)));
typedef float v8f __attribute__((ext_vector_type(8)));

#define BATCH   4
#define NSEQ    2048
#define EDIM    96
#define NHEAD   6
#define HDIM    16
#define QKVDIM  (3 * EDIM)   // 288
#define ROWS    (BATCH * NSEQ) // 8192

static __device__ __forceinline__ v8f wmma_f32_4(v2f a, v2f b, v8f c) {
    // 8 args: (neg_a, A, neg_b, B, c_mod, C, reuse_a, reuse_b)
    return __builtin_amdgcn_wmma_f32_16x16x4_f32(
        false, a, false, b, (short)0, c, false, false);
}

// C[M x N] = A[M x K] @ W[K x N] + bias[N]; one wave per 16x16 tile.
// A-frag (16x4): lane l, vgpr j -> A[row0 + l%16][k0 + (l/16)*2 + j]
// B-frag (4x16): lane l, vgpr j -> W[k0 + (l/16)*2 + j][col0 + l%16]
// D (16x16):     vgpr r, lane l -> C[row0 + r + 8*(l/16)][col0 + l%16]
__global__ __launch_bounds__(256)
void gemm_bias_wmma(const float* __restrict__ A,
                    const float* __restrict__ W,
                    const float* __restrict__ bias,
                    float* __restrict__ C,
                    int M, int K, int N) {
    const int lane = threadIdx.x & 31;
    const int wave = threadIdx.x >> 5;
    const int wavesPerBlock = blockDim.x >> 5;
    const int tileId = blockIdx.x * wavesPerBlock + wave;
    const int tilesN = N >> 4;
    const int tileM = tileId / tilesN;
    const int tileN = tileId - tileM * tilesN;
    if (tileM * 16 >= M) return;            // wave-uniform: EXEC stays all-1s

    const int row0 = tileM << 4;
    const int col0 = tileN << 4;
    const int m    = lane & 15;
    const int sel  = (lane >> 4) << 1;      // 0 or 2
    const int half = (lane >> 4) << 3;      // 0 or 8

    const float* arow = A + (size_t)(row0 + m) * K;

    v8f c = {};
    for (int k0 = 0; k0 < K; k0 += 4) {
        v2f a, b;
        a.x = arow[k0 + sel + 0];
        a.y = arow[k0 + sel + 1];
        b.x = W[(size_t)(k0 + sel + 0) * N + (col0 + m)];
        b.y = W[(size_t)(k0 + sel + 1) * N + (col0 + m)];
        c = wmma_f32_4(a, b, c);
    }

    const float bv = bias[col0 + m];
#pragma unroll
    for (int r = 0; r < 8; ++r) {
        C[(size_t)(row0 + r + half) * N + (col0 + m)] = c[r] + bv;
    }
}

// One block (8 waves) per (b,h).
// Phase 1: S = 0.25 * k^T v   (16x16, reduced over NSEQ via per-wave WMMA + LDS)
// Phase 2: y[b, n, 16h+d] = q @ S
__global__ __launch_bounds__(256)
void attn_kernel(const float* __restrict__ qkv, float* __restrict__ y) {
    const int bh   = blockIdx.x;
    const int b    = bh / NHEAD;
    const int h    = bh - b * NHEAD;
    const int lane = threadIdx.x & 31;
    const int wave = threadIdx.x >> 5;
    const int m    = lane & 15;
    const int sel  = (lane >> 4) << 1;
    const int half = (lane >> 4) << 3;

    const size_t base  = (size_t)b * NSEQ * QKVDIM;
    const float* qbase = qkv + base + 0 * EDIM + HDIM * h;
    const float* kbase = qkv + base + 1 * EDIM + HDIM * h;
    const float* vbase = qkv + base + 2 * EDIM + HDIM * h;

    __shared__ float red[8 * 256];   // 8 partial 16x16 accumulators (D layout)
    __shared__ float S[16 * 16];

    // ---- Phase 1: partial k^T v over this wave's 256 rows ----
    v8f c = {};
    const int nBeg = wave * (NSEQ / 8);
    for (int n0 = nBeg; n0 < nBeg + NSEQ / 8; n0 += 4) {
        const float* krow = kbase + (size_t)(n0 + sel) * QKVDIM;
        const float* vrow = vbase + (size_t)(n0 + sel) * QKVDIM;
        v2f a, bb;
        a.x  = krow[m];               // k[n0+sel  ][m]  -> A[M=m][K=sel]
        a.y  = krow[QKVDIM + m];      // k[n0+sel+1][m]
        bb.x = vrow[m];               // v[n0+sel  ][m]  -> B[K=sel][N=m]
        bb.y = vrow[QKVDIM + m];
        c = wmma_f32_4(a, bb, c);
    }
#pragma unroll
    for (int r = 0; r < 8; ++r) red[wave * 256 + r * 32 + lane] = c[r];
    __syncthreads();

    {   // 256 threads == 256 fragment slots; sum the 8 waves, scale, lay out as S[16][16]
        const int t  = threadIdx.x;
        float s = 0.f;
#pragma unroll
        for (int w = 0; w < 8; ++w) s += red[w * 256 + t];
        const int r  = t >> 5;
        const int ln = t & 31;
        const int row = r + ((ln >> 4) << 3);
        const int col = ln & 15;
        S[row * 16 + col] = s * 0.25f;     // scale = HDIM^-0.5
    }
    __syncthreads();

    // ---- Phase 2: y tiles = q @ S ----
    for (int t = wave; t < NSEQ / 16; t += 8) {
        const int n0 = t << 4;
        const float* qrow = qbase + (size_t)(n0 + m) * QKVDIM;
        v8f acc = {};
#pragma unroll
        for (int k0 = 0; k0 < HDIM; k0 += 4) {
            v2f a, bb;
            a.x  = qrow[k0 + sel + 0];            // A[M=m][K=k0+sel+j]
            a.y  = qrow[k0 + sel + 1];
            bb.x = S[(k0 + sel + 0) * 16 + m];    // B[K][N=m]
            bb.y = S[(k0 + sel + 1) * 16 + m];
            acc = wmma_f32_4(a, bb, acc);
        }
#pragma unroll
        for (int r = 0; r < 8; ++r) {
            y[(size_t)b * NSEQ * EDIM + (size_t)(n0 + r + half) * EDIM + HDIM * h + m] = acc[r];
        }
    }
}

extern "C" void kernel_launch(void* const* d_in, const int* in_sizes, int n_in,
                              void* d_out, int out_size, void* d_ws, size_t ws_size,
                              hipStream_t stream) {
    const float* x    = (const float*)d_in[0];
    const float* Wqkv = (const float*)d_in[1];
    const float* bqkv = (const float*)d_in[2];
    const float* Wff  = (const float*)d_in[3];
    const float* bff  = (const float*)d_in[4];
    float* out = (float*)d_out;

    float* qkv = (float*)d_ws;                       // ROWS x 288  (9.4 MB)
    float* y   = qkv + (size_t)ROWS * QKVDIM;        // ROWS x 96   (3.1 MB)

    // 1) qkv = x @ Wqkv + bqkv : 8192x96 @ 96x288 -> 512*18 = 9216 wave-tiles
    {
        const int tiles  = (ROWS / 16) * (QKVDIM / 16);
        gemm_bias_wmma<<<tiles / 8, 256, 0, stream>>>(x, Wqkv, bqkv, qkv,
                                                      ROWS, EDIM, QKVDIM);
    }

    // 2+3) per (b,h): S = 0.25*k^T v ; y = q @ S
    attn_kernel<<<BATCH * NHEAD, 256, 0, stream>>>(qkv, y);

    // 4) out = y @ Wff + bff : 8192x96 @ 96x96 -> 512*6 = 3072 wave-tiles
    {
        const int tiles = (ROWS / 16) * (EDIM / 16);
        gemm_bias_wmma<<<tiles / 8, 256, 0, stream>>>(y, Wff, bff, out,
                                                      ROWS, EDIM, EDIM);
    }
}